// customRNNforContext_26096221291169
// MI455X (gfx1250) — compile-verified
//
#include <hip/hip_runtime.h>
#include <math.h>

#define T_DIM 2048
#define INP   2048
#define HID   2048
#define OUTD  2048
#define LDW   (INP + HID)   // 4096: row stride of W_i2h / W_i2o

#define KT    32            // K-tile depth (floats)
#define PAD   36            // padded LDS row stride: 36*r mod 64 distinct for r<16

typedef __attribute__((ext_vector_type(2))) float v2f;
typedef __attribute__((ext_vector_type(8))) float v8f;

// ---------------------------------------------------------------------------
// Issue one K-tile's async DMA into LDS (6 b128 ops per thread => ASYNCcnt +6
// per wave). A-tile: 128 rows x 32 floats. W-tile: 64 rows x 32 floats.
// ---------------------------------------------------------------------------
__device__ __forceinline__ void issue_tile(const float* gA, int lda,
                                           const float* gW, int ldw,
                                           const float* ldsA, const float* ldsB,
                                           int tid)
{
#pragma unroll
    for (int i = 0; i < 4; ++i) {               // 1024 chunks / 256 threads
        const int c   = tid + i * 256;
        const int row = c >> 3, seg = c & 7;    // 8 x b128 per 32-float row
        const unsigned dst = (unsigned)(uintptr_t)(ldsA + row * PAD + seg * 4);
        const float*   src = gA + (size_t)row * lda + seg * 4;
        asm volatile("global_load_async_to_lds_b128 %0, %1, off"
                     :: "v"(dst), "v"((unsigned long long)(uintptr_t)src)
                     : "memory");
    }
#pragma unroll
    for (int i = 0; i < 2; ++i) {               // 512 chunks / 256 threads
        const int c   = tid + i * 256;
        const int row = c >> 3, seg = c & 7;
        const unsigned dst = (unsigned)(uintptr_t)(ldsB + row * PAD + seg * 4);
        const float*   src = gW + (size_t)row * ldw + seg * 4;
        asm volatile("global_load_async_to_lds_b128 %0, %1, off"
                     :: "v"(dst), "v"((unsigned long long)(uintptr_t)src)
                     : "memory");
    }
}

// ---------------------------------------------------------------------------
// C[m,n] = act( sum_k A[m,k] * W[n,k] + bias[n] + addend[m,n] )
// Block tile 128(M) x 64(N), 8 waves; per-wave tile 16 x 64 via 4 fp32 WMMAs.
// Double-buffered LDS K-tiles filled by async-to-LDS DMA.
// ---------------------------------------------------------------------------
__global__ __launch_bounds__(256) void gemm_wmma_f32(
    const float* __restrict__ A, int lda,
    const float* __restrict__ W, int ldw,
    const float* __restrict__ bias,              // may be null
    const float* __restrict__ addend, int ldadd, // may be null
    float* __restrict__ C, int ldc,
    int K, int apply_tanh)
{
    __shared__ float lA[2][128 * PAD];
    __shared__ float lB[2][ 64 * PAD];

    const int tid  = threadIdx.x;
    const int lane = tid & 31;
    const int wave = tid >> 5;                   // 0..7
    const int rowBase = blockIdx.y * 128 + wave * 16;
    const int colBase = blockIdx.x * 64;

    // fp32 WMMA fragment layout (ISA 7.12.2): lane l -> row/col (l&15),
    // K-pair (l>>4)*2 .. +1, held as a float2.
    const int ml = lane & 15;
    const int kh = (lane >> 4) << 1;             // 0 or 2

    const float* gA = A + (size_t)(blockIdx.y * 128) * lda;
    const float* gW = W + (size_t)colBase * ldw;

    v8f acc[4];
    acc[0] = v8f{}; acc[1] = v8f{}; acc[2] = v8f{}; acc[3] = v8f{};

    const int nt = K / KT;                       // 64 K-tiles

    // Prologue: DMA tile 0.
    issue_tile(gA, lda, gW, ldw, lA[0], lB[0], tid);

#pragma unroll 1
    for (int t = 0; t < nt; ++t) {
        // DMA the next tile into the other buffer, then wait only for the
        // current tile's batch (6 ops) to land.
        if (t + 1 < nt) {
            issue_tile(gA + (t + 1) * KT, lda, gW + (t + 1) * KT, ldw,
                       lA[(t + 1) & 1], lB[(t + 1) & 1], tid);
            asm volatile("s_wait_asynccnt 0x6" ::: "memory");
        } else {
            asm volatile("s_wait_asynccnt 0x0" ::: "memory");
        }
        __syncthreads();   // current tile visible to all waves

        const float* pA  = &lA[t & 1][(wave * 16 + ml) * PAD + kh];
        const float* pB0 = &lB[t & 1][( 0 + ml) * PAD + kh];
        const float* pB1 = &lB[t & 1][(16 + ml) * PAD + kh];
        const float* pB2 = &lB[t & 1][(32 + ml) * PAD + kh];
        const float* pB3 = &lB[t & 1][(48 + ml) * PAD + kh];

        // Pipelined fragment reads: prefetch step s+1 before WMMAs of step s.
        v2f a_c  = *(const v2f*)(pA);
        v2f b0_c = *(const v2f*)(pB0);
        v2f b1_c = *(const v2f*)(pB1);
        v2f b2_c = *(const v2f*)(pB2);
        v2f b3_c = *(const v2f*)(pB3);
#pragma unroll
        for (int s = 0; s < KT / 4; ++s) {
            v2f a_n = a_c, b0_n = b0_c, b1_n = b1_c, b2_n = b2_c, b3_n = b3_c;
            if (s + 1 < KT / 4) {
                a_n  = *(const v2f*)(pA  + (s + 1) * 4);
                b0_n = *(const v2f*)(pB0 + (s + 1) * 4);
                b1_n = *(const v2f*)(pB1 + (s + 1) * 4);
                b2_n = *(const v2f*)(pB2 + (s + 1) * 4);
                b3_n = *(const v2f*)(pB3 + (s + 1) * 4);
            }
            acc[0] = __builtin_amdgcn_wmma_f32_16x16x4_f32(false, a_c, false, b0_c, (short)0, acc[0], false, false);
            acc[1] = __builtin_amdgcn_wmma_f32_16x16x4_f32(false, a_c, false, b1_c, (short)0, acc[1], false, false);
            acc[2] = __builtin_amdgcn_wmma_f32_16x16x4_f32(false, a_c, false, b2_c, (short)0, acc[2], false, false);
            acc[3] = __builtin_amdgcn_wmma_f32_16x16x4_f32(false, a_c, false, b3_c, (short)0, acc[3], false, false);
            a_c = a_n; b0_c = b0_n; b1_c = b1_n; b2_c = b2_n; b3_c = b3_n;
        }
        __syncthreads();   // tile consumed; its buffer may be overwritten
    }

    // C/D layout: VGPR p, lanes 0-15 -> M=p, lanes 16-31 -> M=p+8; N = lane&15
    const int rsel = lane >> 4;
#pragma unroll
    for (int j = 0; j < 4; ++j) {
        const int col = colBase + j * 16 + ml;
        const float b = bias ? bias[col] : 0.0f;
#pragma unroll
        for (int p = 0; p < 8; ++p) {
            const int row = rowBase + p + 8 * rsel;
            float v = acc[j][p] + b;
            if (addend) v += addend[(size_t)row * ldadd + col];
            if (apply_tanh) v = tanhf(v);
            C[(size_t)row * ldc + col] = v;
        }
    }
}

// ---------------------------------------------------------------------------
// Device-wide barrier: monotonic counter in global memory (zeroed by the
// graph's memset node). All 256 blocks are resident (persistent kernel).
// ---------------------------------------------------------------------------
__device__ __forceinline__ void grid_sync(unsigned* ctr, unsigned target)
{
    __syncthreads();
    if (threadIdx.x == 0) {
        __threadfence();
        __hip_atomic_fetch_add(ctr, 1u, __ATOMIC_ACQ_REL, __HIP_MEMORY_SCOPE_AGENT);
        while (__hip_atomic_load(ctr, __ATOMIC_ACQUIRE, __HIP_MEMORY_SCOPE_AGENT) < target) {
            __builtin_amdgcn_s_sleep(2);
        }
    }
    __syncthreads();
}

// ---------------------------------------------------------------------------
// Persistent linear-recurrence scan:
//   H[0] = 0 ;  H[t] = XH[t-1] + Whh * H[t-1],  t = 1..T-1
// 256 blocks x 8 waves -> one wave per hidden row. Whh (16MB) stays L2-resident.
// ---------------------------------------------------------------------------
__global__ __launch_bounds__(256) void rnn_scan(
    const float* __restrict__ XH,   // T x HID
    const float* __restrict__ Whh,  // HID rows, stride LDW (col-offset by INP)
    float* __restrict__ H,          // T x HID
    unsigned* __restrict__ ctr)
{
    const int lane = threadIdx.x & 31;
    const int wave = threadIdx.x >> 5;
    const int row  = blockIdx.x * 8 + wave;   // 256*8 = 2048 rows
    const unsigned nb = gridDim.x;

    const int gid = blockIdx.x * blockDim.x + threadIdx.x;
    if (gid < HID) H[gid] = 0.0f;             // H[0] = 0
    grid_sync(ctr, nb);

    const float4* wrow = (const float4*)(Whh + (size_t)row * LDW);

    for (int t = 1; t < T_DIM; ++t) {
        const float4* hprev = (const float4*)(H + (size_t)(t - 1) * HID);
        float acc = 0.0f;
#pragma unroll 4
        for (int k = lane; k < HID / 4; k += 32) {   // coalesced b128 loads
            float4 w = wrow[k];
            float4 h = hprev[k];
            acc += w.x * h.x + w.y * h.y + w.z * h.z + w.w * h.w;
        }
#pragma unroll
        for (int off = 16; off > 0; off >>= 1)
            acc += __shfl_down(acc, off, 32);
        if (lane == 0)
            H[(size_t)t * HID + row] = acc + XH[(size_t)(t - 1) * HID + row];
        grid_sync(ctr, nb * (unsigned)(t + 1));
    }
}

// ---------------------------------------------------------------------------
extern "C" void kernel_launch(void* const* d_in, const int* in_sizes, int n_in,
                              void* d_out, int out_size, void* d_ws, size_t ws_size,
                              hipStream_t stream)
{
    (void)in_sizes; (void)n_in; (void)out_size; (void)ws_size;

    const float* X   = (const float*)d_in[0];  // T x INP
    const float* Wih = (const float*)d_in[1];  // HID x (INP+HID)
    const float* bih = (const float*)d_in[2];  // HID
    const float* Wio = (const float*)d_in[3];  // OUT x (INP+HID)
    const float* bio = (const float*)d_in[4];  // OUT
    float* out = (float*)d_out;                // T x OUT (fp32)

    const size_t MAT = (size_t)T_DIM * HID * sizeof(float);  // 16 MB
    char* ws = (char*)d_ws;
    float*    XH  = (float*)(ws);
    float*    XO  = (float*)(ws + MAT);
    float*    H   = (float*)(ws + 2 * MAT);
    unsigned* ctr = (unsigned*)(ws + 3 * MAT);

    hipMemsetAsync(ctr, 0, 256, stream);

    dim3 blk(256);
    dim3 grd(OUTD / 64, T_DIM / 128);   // 32 x 16 blocks

    // XH = X * Wxh^T + b_i2h
    gemm_wmma_f32<<<grd, blk, 0, stream>>>(X, INP, Wih, LDW, bih,
                                           nullptr, 0, XH, HID, INP, 0);
    // XO = X * Wxo^T + b_i2o
    gemm_wmma_f32<<<grd, blk, 0, stream>>>(X, INP, Wio, LDW, bio,
                                           nullptr, 0, XO, OUTD, INP, 0);
    // Sequential linear scan -> H
    rnn_scan<<<dim3(HID / 8), blk, 0, stream>>>(XH, Wih + INP, H, ctr);
    // out = tanh(XO + H * Who^T)
    gemm_wmma_f32<<<grd, blk, 0, stream>>>(H, HID, Wio + INP, LDW, nullptr,
                                           XO, OUTD, out, OUTD, HID, 1);
}